// PointFusion_36919538876793
// MI455X (gfx1250) — compile-verified
//
#include <hip/hip_runtime.h>
#include <hip/hip_bf16.h>
#include <stdint.h>

// ---------------------------------------------------------------------------
// PointFusion for MI455X (gfx1250): f32 WMMA implicit-GEMM conv + fused
// projection/bilinear-gather GEMM + BatchNorm(train) + ReLU.
// Round 4: software-pipelined conv (double-buffered LDS, async B copies and
// register-staged A loads overlapped with WMMA).
// ---------------------------------------------------------------------------

typedef float v2f __attribute__((ext_vector_type(2)));
typedef float v4f __attribute__((ext_vector_type(4)));
typedef float v8f __attribute__((ext_vector_type(8)));
typedef int   v4i __attribute__((ext_vector_type(4)));

typedef __attribute__((address_space(1))) v4i* g4p;   // global int4*
typedef __attribute__((address_space(3))) v4i* l4p;   // LDS int4*

constexpr int cNLVL = 3;
constexpr int cNCAM = 5;
constexpr int cNPTS = 65536;
constexpr int cCIN  = 256;
constexpr int cMID  = 128;
constexpr int cOUT  = 128;
constexpr int cPTSC = 128;
constexpr int cH    = 64;
constexpr int cW    = 112;
constexpr int cHW   = cH * cW;          // 7168
constexpr int cKCONV = cCIN * 9;        // 2304
constexpr int cKIMG  = cMID * cNLVL;    // 384
constexpr int cNROWS = 2 * cNPTS;       // 131072
constexpr int cNIMG  = 2 * cNCAM;       // 10
constexpr int cAS    = 36;              // fuse/final A-tile row stride (dwords)
constexpr int cASZ   = 64 * 33;         // conv A buffer size (dwords)
constexpr int cBSZ   = 32 * 128;        // conv B buffer size (dwords)
constexpr int cNCHUNK = cKCONV / 32;    // 72

// workspace layout (float offsets)
constexpr size_t SZ_CONVOUT  = (size_t)cNLVL * cNIMG * cHW * cMID;   // 27,525,120
constexpr size_t OFF_WTC     = SZ_CONVOUT;
constexpr size_t SZ_WTC      = (size_t)cNLVL * cKCONV * cMID;        // 884,736
constexpr size_t OFF_WTI     = OFF_WTC + SZ_WTC;
constexpr size_t SZ_WTI      = (size_t)cKIMG * cOUT;
constexpr size_t OFF_WTP     = OFF_WTI + SZ_WTI;
constexpr size_t SZ_WTP      = (size_t)cPTSC * cOUT;
constexpr size_t OFF_STATS   = OFF_WTP + SZ_WTP;                     // 512 floats

#if __has_builtin(__builtin_amdgcn_global_load_async_to_lds_b128)
#define PF_ASYNC 1
#else
#define PF_ASYNC 0
#endif

__device__ __forceinline__ void cp_lds_b128(float* lds, const float* g) {
#if PF_ASYNC
  __builtin_amdgcn_global_load_async_to_lds_b128(
      (g4p)(uintptr_t)g, (l4p)(uintptr_t)lds, 0, 0);
#else
  *(v4f*)lds = *(const v4f*)g;
#endif
}

__device__ __forceinline__ void wait_async_copies() {
#if PF_ASYNC
#if __has_builtin(__builtin_amdgcn_s_wait_asynccnt)
  __builtin_amdgcn_s_wait_asynccnt(0);
#else
  asm volatile("s_wait_asynccnt 0x0" ::: "memory");
#endif
#endif
}

__device__ __forceinline__ v8f wmma_f32(v2f a, v2f b, v8f c) {
  return __builtin_amdgcn_wmma_f32_16x16x4_f32(false, a, false, b, (short)0, c,
                                               false, false);
}

__device__ __forceinline__ void zero8(v8f* acc, int n) {
  for (int t = 0; t < n; ++t)
    for (int j = 0; j < 8; ++j) acc[t][j] = 0.0f;
}

// ---------------------------------------------------------------------------
// Kernel 0: transpose weights to K-major [K][N]; zero stat accumulators.
// ---------------------------------------------------------------------------
__global__ void pf_prep_kernel(const float* __restrict__ conv_w,
                               const float* __restrict__ img_w,
                               const float* __restrict__ pts_w,
                               float* __restrict__ wTc,
                               float* __restrict__ wTi,
                               float* __restrict__ wTp,
                               float* __restrict__ stats) {
  int idx = blockIdx.x * blockDim.x + threadIdx.x;
  int stride = gridDim.x * blockDim.x;
  for (int i = idx; i < cNLVL * cKCONV * cMID; i += stride) {
    int lvl = i / (cKCONV * cMID);
    int rem = i % (cKCONV * cMID);
    int k = rem / cMID, n = rem % cMID;
    wTc[i] = conv_w[(size_t)(lvl * cMID + n) * cKCONV + k];
  }
  for (int i = idx; i < cKIMG * cOUT; i += stride) {
    int k = i / cOUT, n = i % cOUT;
    wTi[i] = img_w[(size_t)n * cKIMG + k];
  }
  for (int i = idx; i < cPTSC * cOUT; i += stride) {
    int k = i / cOUT, n = i % cOUT;
    wTp[i] = pts_w[(size_t)n * cPTSC + k];
  }
  for (int i = idx; i < 512; i += stride) stats[i] = 0.0f;
}

// ---------------------------------------------------------------------------
// Kernel 1: 3x3 SAME conv as software-pipelined implicit GEMM.
//   grid = (2 x-tiles, 64 rows, 30 lvl*img), block = 256 (8 waves).
//   M = 64 output positions, N = 128 out-channels, K = 2304.
//   Double-buffered LDS: async B copy + register-staged A overlap with WMMA.
// ---------------------------------------------------------------------------
__global__ __launch_bounds__(256) void pf_conv_kernel(
    const float* __restrict__ img_feats, const float* __restrict__ wTc,
    const float* __restrict__ conv_b, float* __restrict__ conv_out) {
  __shared__ __align__(16) float As[2 * cASZ];
  __shared__ __align__(16) float Bs[2 * cBSZ];
  __shared__ int tbl[cKCONV];   // (cin*HW)<<4 | ky<<2 | kx

  const int tid = threadIdx.x;
  const int xb  = blockIdx.x * 64;
  const int y   = blockIdx.y;
  const int li  = blockIdx.z;          // lvl*10 + img
  const int lvl = li / cNIMG;

  const float* in = img_feats + (size_t)li * cCIN * cHW;
  const float* wt = wTc + (size_t)lvl * cKCONV * cMID;

  // build packed k -> (cin*HW, ky, kx) table once per block
  for (int k = tid; k < cKCONV; k += 256) {
    int cin = k / 9, r9 = k % 9;
    int ky = r9 / 3, kx = r9 % 3;
    tbl[k] = ((cin * cHW) << 4) | (ky << 2) | kx;
  }

  const int lane = tid & 31;
  const int wid  = tid >> 5;
  const int mt   = (wid & 3) * 16;     // M-tile base (positions)
  const int n0   = (wid >> 2) * 64;    // N half base (channels)
  const int half = lane >> 4;
  const int l16  = lane & 15;
  const int ybase = y * cW + xb;

  v8f acc[4];
  zero8(acc, 4);
  __syncthreads();   // table ready

  // ---- prologue: stage chunk 0 into buffer 0 ----
#pragma unroll
  for (int j = 0; j < 4; ++j) {
    int e = tid + j * 256;
    int kk = e >> 5, n4 = (e & 31) * 4;
    cp_lds_b128(&Bs[kk * 128 + n4], &wt[kk * cMID + n4]);
  }
#pragma unroll
  for (int j = 0; j < 8; ++j) {
    int e = tid + j * 256;
    int m = e & 63, kk = e >> 6;
    int t = tbl[kk];
    int kx = t & 3, ky = (t >> 2) & 3;
    int xx = xb + m + kx - 1, yy = y + ky - 1;
    float v = 0.0f;
    if (xx >= 0 && xx < cW && yy >= 0 && yy < cH)
      v = in[(t >> 4) + (ky - 1) * cW + (kx - 1) + ybase + m];
    As[m * 33 + kk] = v;
  }
  wait_async_copies();
  __syncthreads();

  int cur = 0;
  for (int kc = 0; kc < cNCHUNK; ++kc) {
    const int nxt = cur ^ 1;
    const bool more = (kc + 1) < cNCHUNK;
    float areg[8];
    int   aoff[8];
    if (more) {
      const int kb = (kc + 1) * 32;
      // issue next B tile on the async engine (no VGPR traffic)
#pragma unroll
      for (int j = 0; j < 4; ++j) {
        int e = tid + j * 256;
        int kk = e >> 5, n4 = (e & 31) * 4;
        cp_lds_b128(&Bs[nxt * cBSZ + kk * 128 + n4],
                    &wt[(kb + kk) * cMID + n4]);
      }
      // issue next A tile loads; values land in registers during the WMMAs
#pragma unroll
      for (int j = 0; j < 8; ++j) {
        int e = tid + j * 256;
        int m = e & 63, kk = e >> 6;
        int t = tbl[kb + kk];
        int kx = t & 3, ky = (t >> 2) & 3;
        int xx = xb + m + kx - 1, yy = y + ky - 1;
        float v = 0.0f;
        if (xx >= 0 && xx < cW && yy >= 0 && yy < cH)
          v = in[(t >> 4) + (ky - 1) * cW + (kx - 1) + ybase + m];
        areg[j] = v;
        aoff[j] = nxt * cASZ + m * 33 + kk;
      }
    }
    // ---- 32 WMMAs on the current buffers ----
    const float* Ac = &As[cur * cASZ];
    const float* Bc = &Bs[cur * cBSZ];
    for (int kk = 0; kk < 32; kk += 4) {
      v2f a;
      a.x = Ac[(mt + l16) * 33 + kk + 2 * half + 0];
      a.y = Ac[(mt + l16) * 33 + kk + 2 * half + 1];
#pragma unroll
      for (int nt = 0; nt < 4; ++nt) {
        v2f b;
        b.x = Bc[(kk + 2 * half + 0) * 128 + n0 + nt * 16 + l16];
        b.y = Bc[(kk + 2 * half + 1) * 128 + n0 + nt * 16 + l16];
        acc[nt] = wmma_f32(a, b, acc[nt]);
      }
    }
    // drain: park staged A values into the next LDS buffer
    if (more) {
#pragma unroll
      for (int j = 0; j < 8; ++j) As[aoff[j]] = areg[j];
    }
    wait_async_copies();
    __syncthreads();
    cur = nxt;
  }

  // bias + store (channel-last)
#pragma unroll
  for (int nt = 0; nt < 4; ++nt) {
    int n = n0 + nt * 16 + l16;
    float bias = conv_b[lvl * cMID + n];
#pragma unroll
    for (int r = 0; r < 8; ++r) {
      int m = mt + r + 8 * half;
      int x = xb + m;
      if (x < cW)
        conv_out[((size_t)li * cHW + y * cW + x) * cMID + n] = acc[nt][r] + bias;
    }
  }
}

// ---------------------------------------------------------------------------
// Kernel 2: project + bilinear gather + img GEMM (store img_lin to d_out)
//           + pts GEMM (stats only). block = 128 points, 256 threads.
// ---------------------------------------------------------------------------
__global__ __launch_bounds__(256) void pf_fuse_kernel(
    const float* __restrict__ conv_out, const float* __restrict__ pts_xyz,
    const float* __restrict__ pts_feats, const float* __restrict__ Tmat,
    const float* __restrict__ Kmat, const int* __restrict__ cam_ids,
    const float* __restrict__ wTi, const float* __restrict__ img_b,
    const float* __restrict__ wTp, const float* __restrict__ pts_b,
    float* __restrict__ stats, float* __restrict__ out_lin) {
  __shared__ __align__(16) float As[128 * cAS];
  __shared__ __align__(16) float Bs[32 * 128];
  __shared__ float cw[128][4];
  __shared__ int   coff[128][4];
  __shared__ float s_is[128], s_iq[128], s_ps[128], s_pq[128];

  const int tid = threadIdx.x;
  const int blk = blockIdx.x;

  if (tid < 128) {
    s_is[tid] = 0.0f; s_iq[tid] = 0.0f; s_ps[tid] = 0.0f; s_pq[tid] = 0.0f;
    const int p  = tid;
    const int gi = blk * 128 + p;       // global point row
    const int bi = gi / cNPTS;          // batch
    const float* xyz = pts_xyz + (size_t)gi * 3;
    const float X = xyz[0], Y = xyz[1], Z = xyz[2];
    const int cam = cam_ids[gi];
    const float* Tc = Tmat + cam * 16;
    const float* Kc = Kmat + cam * 9;
    float pcx = Tc[0] * X + Tc[1] * Y + Tc[2]  * Z + Tc[3];
    float pcy = Tc[4] * X + Tc[5] * Y + Tc[6]  * Z + Tc[7];
    float pcz = Tc[8] * X + Tc[9] * Y + Tc[10] * Z + Tc[11];
    float fx = Kc[0] * 0.125f, sk = Kc[1], cx = Kc[2] * 0.125f;
    float k10 = Kc[3], fy = Kc[4] * 0.125f, cy = Kc[5] * 0.125f;
    float pix = fx * pcx + sk * pcy + cx * pcz;
    float piy = k10 * pcx + fy * pcy + cy * pcz;
    float piz = Kc[6] * pcx + Kc[7] * pcy + Kc[8] * pcz;
    float u = pix / piz, v = piy / piz;
    float gx = u / (float)cW * 2.0f - 1.0f;
    float gy = v / (float)cH * 2.0f - 1.0f;
    float px = (gx + 1.0f) * 0.5f * (float)(cW - 1);  // align_corners=True
    float py = (gy + 1.0f) * 0.5f * (float)(cH - 1);
    float x0 = floorf(px), y0 = floorf(py);
    float wx1 = px - x0, wy1 = py - y0;
    const int img = bi * cNCAM + cam;
#pragma unroll
    for (int c = 0; c < 4; ++c) {
      float xf = x0 + (float)(c & 1);
      float yf = y0 + (float)(c >> 1);
      bool valid = (xf >= 0.0f) && (xf <= (float)(cW - 1)) &&
                   (yf >= 0.0f) && (yf <= (float)(cH - 1));
      int xi = (int)fminf(fmaxf(xf, 0.0f), (float)(cW - 1));
      int yi = (int)fminf(fmaxf(yf, 0.0f), (float)(cH - 1));
      float wx = (c & 1) ? wx1 : (1.0f - wx1);
      float wy = (c >> 1) ? wy1 : (1.0f - wy1);
      cw[p][c]   = wx * wy * (valid ? 1.0f : 0.0f);
      coff[p][c] = (img * cHW + yi * cW + xi) * cMID;
    }
  }
  __syncthreads();

  const int lane = tid & 31, wid = tid >> 5;
  const int half = lane >> 4, l16 = lane & 15;
  const int mt = wid * 16;   // 8 waves x 16 points = 128 points

  // ----- image branch: [128 pts x 384] @ [384 x 128] -----
  v8f acc[8];
  zero8(acc, 8);
  for (int kc = 0; kc < cKIMG / 32; ++kc) {
    for (int e = tid; e < 32 * 32; e += 256) {   // async B stage
      int kk = e >> 5, n4 = (e & 31) * 4;
      cp_lds_b128(&Bs[kk * 128 + n4], &wTi[(kc * 32 + kk) * cOUT + n4]);
    }
    for (int e = tid; e < 128 * 32; e += 256) {  // gathered A stage
      int pt = e >> 5, kk = e & 31;
      int k = kc * 32 + kk;
      int lvl = k >> 7, ch = k & 127;
      const float* cbase = conv_out + (size_t)lvl * (cNIMG * (size_t)cHW * cMID);
      float v = cw[pt][0] * cbase[coff[pt][0] + ch]
              + cw[pt][1] * cbase[coff[pt][1] + ch]
              + cw[pt][2] * cbase[coff[pt][2] + ch]
              + cw[pt][3] * cbase[coff[pt][3] + ch];
      As[pt * cAS + kk] = v;
    }
    wait_async_copies();
    __syncthreads();
    for (int kk = 0; kk < 32; kk += 4) {
      v2f a;
      a.x = As[(mt + l16) * cAS + kk + 2 * half + 0];
      a.y = As[(mt + l16) * cAS + kk + 2 * half + 1];
#pragma unroll
      for (int nt = 0; nt < 8; ++nt) {
        v2f b;
        b.x = Bs[(kk + 2 * half + 0) * 128 + nt * 16 + l16];
        b.y = Bs[(kk + 2 * half + 1) * 128 + nt * 16 + l16];
        acc[nt] = wmma_f32(a, b, acc[nt]);
      }
    }
    __syncthreads();
  }
#pragma unroll
  for (int nt = 0; nt < 8; ++nt) {
    int n = nt * 16 + l16;
    float bias = img_b[n];
    float s = 0.0f, q = 0.0f;
#pragma unroll
    for (int r = 0; r < 8; ++r) {
      float v = acc[nt][r] + bias;
      int row = blk * 128 + mt + r + 8 * half;
      out_lin[(size_t)row * cOUT + n] = v;   // pre-BN img_lin parked in d_out
      s += v; q += v * v;
    }
    atomicAdd(&s_is[n], s);
    atomicAdd(&s_iq[n], q);
  }

  // ----- points branch: [128 pts x 128] @ [128 x 128] (stats only) -----
  zero8(acc, 8);
  for (int kc = 0; kc < cPTSC / 32; ++kc) {
    for (int e = tid; e < 128 * 8; e += 256) {   // async A stage (pts rows)
      int pt = e >> 3, q4 = (e & 7) * 4;
      cp_lds_b128(&As[pt * cAS + q4],
                  &pts_feats[(size_t)(blk * 128 + pt) * cPTSC + kc * 32 + q4]);
    }
    for (int e = tid; e < 32 * 32; e += 256) {   // async B stage
      int kk = e >> 5, n4 = (e & 31) * 4;
      cp_lds_b128(&Bs[kk * 128 + n4], &wTp[(kc * 32 + kk) * cOUT + n4]);
    }
    wait_async_copies();
    __syncthreads();
    for (int kk = 0; kk < 32; kk += 4) {
      v2f a;
      a.x = As[(mt + l16) * cAS + kk + 2 * half + 0];
      a.y = As[(mt + l16) * cAS + kk + 2 * half + 1];
#pragma unroll
      for (int nt = 0; nt < 8; ++nt) {
        v2f b;
        b.x = Bs[(kk + 2 * half + 0) * 128 + nt * 16 + l16];
        b.y = Bs[(kk + 2 * half + 1) * 128 + nt * 16 + l16];
        acc[nt] = wmma_f32(a, b, acc[nt]);
      }
    }
    __syncthreads();
  }
#pragma unroll
  for (int nt = 0; nt < 8; ++nt) {
    int n = nt * 16 + l16;
    float bias = pts_b[n];
    float s = 0.0f, q = 0.0f;
#pragma unroll
    for (int r = 0; r < 8; ++r) {
      float v = acc[nt][r] + bias;
      s += v; q += v * v;
    }
    atomicAdd(&s_ps[n], s);
    atomicAdd(&s_pq[n], q);
  }
  __syncthreads();
  if (tid < 128) {
    atomicAdd(&stats[tid],       s_is[tid]);
    atomicAdd(&stats[128 + tid], s_iq[tid]);
    atomicAdd(&stats[256 + tid], s_ps[tid]);
    atomicAdd(&stats[384 + tid], s_pq[tid]);
  }
}

// ---------------------------------------------------------------------------
// Kernel 3: recompute pts GEMM, apply both BatchNorms + ReLU into d_out.
// ---------------------------------------------------------------------------
__global__ __launch_bounds__(256) void pf_final_kernel(
    const float* __restrict__ pts_feats, const float* __restrict__ wTp,
    const float* __restrict__ pts_b, const float* __restrict__ img_g,
    const float* __restrict__ img_be, const float* __restrict__ pts_g,
    const float* __restrict__ pts_be, const float* __restrict__ stats,
    float* __restrict__ out) {
  __shared__ __align__(16) float As[128 * cAS];
  __shared__ __align__(16) float Bs[32 * 128];

  const int tid = threadIdx.x;
  const int blk = blockIdx.x;
  const int lane = tid & 31, wid = tid >> 5;
  const int half = lane >> 4, l16 = lane & 15;
  const int mt = wid * 16;

  v8f acc[8];
  zero8(acc, 8);
  for (int kc = 0; kc < cPTSC / 32; ++kc) {
    for (int e = tid; e < 128 * 8; e += 256) {
      int pt = e >> 3, q4 = (e & 7) * 4;
      cp_lds_b128(&As[pt * cAS + q4],
                  &pts_feats[(size_t)(blk * 128 + pt) * cPTSC + kc * 32 + q4]);
    }
    for (int e = tid; e < 32 * 32; e += 256) {
      int kk = e >> 5, n4 = (e & 31) * 4;
      cp_lds_b128(&Bs[kk * 128 + n4], &wTp[(kc * 32 + kk) * cOUT + n4]);
    }
    wait_async_copies();
    __syncthreads();
    for (int kk = 0; kk < 32; kk += 4) {
      v2f a;
      a.x = As[(mt + l16) * cAS + kk + 2 * half + 0];
      a.y = As[(mt + l16) * cAS + kk + 2 * half + 1];
#pragma unroll
      for (int nt = 0; nt < 8; ++nt) {
        v2f b;
        b.x = Bs[(kk + 2 * half + 0) * 128 + nt * 16 + l16];
        b.y = Bs[(kk + 2 * half + 1) * 128 + nt * 16 + l16];
        acc[nt] = wmma_f32(a, b, acc[nt]);
      }
    }
    __syncthreads();
  }

  const float inv = 1.0f / (float)cNROWS;
#pragma unroll
  for (int nt = 0; nt < 8; ++nt) {
    int n = nt * 16 + l16;
    float is = stats[n],       iq = stats[128 + n];
    float ps = stats[256 + n], pq = stats[384 + n];
    float imu = is * inv, ivar = iq * inv - imu * imu;
    float pmu = ps * inv, pvar = pq * inv - pmu * pmu;
    float isc = rsqrtf(ivar + 1e-3f) * img_g[n];
    float psc = rsqrtf(pvar + 1e-3f) * pts_g[n];
    float ib = img_be[n], pb = pts_be[n];
    float pbias = pts_b[n];
#pragma unroll
    for (int r = 0; r < 8; ++r) {
      int row = blk * 128 + mt + r + 8 * half;
      float iv = out[(size_t)row * cOUT + n];      // img_lin from pass 1
      float pv = acc[nt][r] + pbias;               // pts_lin recomputed
      float val = (iv - imu) * isc + ib + (pv - pmu) * psc + pb;
      out[(size_t)row * cOUT + n] = fmaxf(val, 0.0f);
    }
  }
}

// ---------------------------------------------------------------------------
extern "C" void kernel_launch(void* const* d_in, const int* in_sizes, int n_in,
                              void* d_out, int out_size, void* d_ws,
                              size_t ws_size, hipStream_t stream) {
  const float* img_feats = (const float*)d_in[0];
  const float* pts_xyz   = (const float*)d_in[1];
  const float* pts_feats = (const float*)d_in[2];
  const float* Tm        = (const float*)d_in[3];
  const float* Km        = (const float*)d_in[4];
  const float* conv_w    = (const float*)d_in[5];
  const float* conv_b    = (const float*)d_in[6];
  const float* img_w     = (const float*)d_in[7];
  const float* img_b     = (const float*)d_in[8];
  const float* img_g     = (const float*)d_in[9];
  const float* img_be    = (const float*)d_in[10];
  const float* pts_w     = (const float*)d_in[11];
  const float* pts_b     = (const float*)d_in[12];
  const float* pts_g     = (const float*)d_in[13];
  const float* pts_be    = (const float*)d_in[14];
  const int*   cam_ids   = (const int*)d_in[15];

  float* out = (float*)d_out;
  float* ws  = (float*)d_ws;
  float* conv_out = ws;
  float* wTc   = ws + OFF_WTC;
  float* wTi   = ws + OFF_WTI;
  float* wTp   = ws + OFF_WTP;
  float* stats = ws + OFF_STATS;

  pf_prep_kernel<<<256, 256, 0, stream>>>(conv_w, img_w, pts_w, wTc, wTi, wTp,
                                          stats);
  pf_conv_kernel<<<dim3(2, cH, cNLVL * cNIMG), 256, 0, stream>>>(
      img_feats, wTc, conv_b, conv_out);
  pf_fuse_kernel<<<cNROWS / 128, 256, 0, stream>>>(
      conv_out, pts_xyz, pts_feats, Tm, Km, cam_ids, wTi, img_b, wTp, pts_b,
      stats, out);
  pf_final_kernel<<<cNROWS / 128, 256, 0, stream>>>(
      pts_feats, wTp, pts_b, img_g, img_be, pts_g, pts_be, stats, out);
}